// Syntax_Transformer_15779709845998
// MI455X (gfx1250) — compile-verified
//
#include <hip/hip_runtime.h>

typedef __attribute__((ext_vector_type(2))) float v2f;
typedef __attribute__((ext_vector_type(8))) float v8f;
typedef __attribute__((ext_vector_type(4))) unsigned v4u;
typedef __attribute__((ext_vector_type(4))) int v4i;
typedef __attribute__((ext_vector_type(8))) int v8i;

#define B_ 8
#define S_ 160
#define H_ 768
#define NH_ 12
#define DH_ 64
#define DEP_ 64
#define M_ (B_*S_)      // 1280
#define LDSROW_ 65      // 64 dep dwords + 1 pad dword (TDM pad) -> bank-conflict-free

__device__ __forceinline__ v8f wmma4(float ax, float ay, float bx, float by, v8f c) {
    v2f a = {ax, ay};
    v2f b = {bx, by};
    return __builtin_amdgcn_wmma_f32_16x16x4_f32(false, a, false, b, (short)0, c,
                                                 false, false);
}

// ---------------- Kernel 1: q/k/v = X @ W + b  (M=1280, N=768, K=768) ----------
__global__ void qkv_kernel(const float* __restrict__ X,
                           const float* __restrict__ Wq, const float* __restrict__ bq,
                           const float* __restrict__ Wk, const float* __restrict__ bk,
                           const float* __restrict__ Wv, const float* __restrict__ bv,
                           float* __restrict__ q, float* __restrict__ k,
                           float* __restrict__ v) {
    const float* W; const float* bias; float* O;
    if (blockIdx.z == 0)      { W = Wq; bias = bq; O = q; }
    else if (blockIdx.z == 1) { W = Wk; bias = bk; O = k; }
    else                      { W = Wv; bias = bv; O = v; }
    const int wave = threadIdx.x >> 5;
    const int lane = threadIdx.x & 31;
    const int half = lane >> 4, l = lane & 15;
    const int m0 = blockIdx.x * 16;
    const int n0 = blockIdx.y * 64 + wave * 16;
    const float* Arow = X + (size_t)(m0 + l) * H_;
    v8f acc = {0,0,0,0,0,0,0,0};
    for (int kk = 0; kk < H_; kk += 4) {
        const int ka = kk + 2 * half;
        acc = wmma4(Arow[ka], Arow[ka + 1],
                    W[(size_t)ka * H_ + n0 + l], W[(size_t)(ka + 1) * H_ + n0 + l],
                    acc);
    }
    const int col = n0 + l;
    const float bb = bias[col];
#pragma unroll
    for (int r = 0; r < 8; ++r)
        O[(size_t)(m0 + half * 8 + r) * H_ + col] = acc[r] + bb;
}

// ---------------- Kernel 2: qe[m, h, dep] = q_h[m,:] @ Wek_h^T  ---------------
__global__ void qe_kernel(const float* __restrict__ q, const float* __restrict__ Wek,
                          float* __restrict__ qe) {
    const int lane = threadIdx.x & 31;
    const int half = lane >> 4, l = lane & 15;
    const int m0 = blockIdx.x * 16;
    const int n0 = blockIdx.y * 16;
    const int h  = blockIdx.z;
    const float* Arow = q + (size_t)(m0 + l) * H_ + h * DH_;
    const float* Bcol = Wek + (size_t)(n0 + l) * H_ + h * DH_; // B[k,n]=Wek[n, h*64+k]
    v8f acc = {0,0,0,0,0,0,0,0};
    for (int kk = 0; kk < DH_; kk += 4) {
        const int ka = kk + 2 * half;
        acc = wmma4(Arow[ka], Arow[ka + 1], Bcol[ka], Bcol[ka + 1], acc);
    }
#pragma unroll
    for (int r = 0; r < 8; ++r)
        qe[(size_t)(m0 + half * 8 + r) * (16 * DEP_) + h * DEP_ + n0 + l] = acc[r];
}

// ---------------- Kernel 2b: qdb[m,h] = q_h[m,:] . bek_h ----------------------
__global__ void qbek_kernel(const float* __restrict__ q, const float* __restrict__ bek,
                            float* __restrict__ qdb) {
    const int m = blockIdx.x, h = blockIdx.y, lane = threadIdx.x;
    const float* qr = q + (size_t)m * H_ + h * DH_;
    const float* br = bek + h * DH_;
    float s = qr[lane] * br[lane] + qr[lane + 32] * br[lane + 32];
    for (int off = 16; off; off >>= 1) s += __shfl_xor(s, off, 32);
    if (lane == 0) qdb[(size_t)m * NH_ + h] = s;
}

// ---------------- Kernel 3a: s_qk[b,h,i,j] = q_i . k_j  (per (b,h): 160x160x64)
__global__ void sqk_kernel(const float* __restrict__ q, const float* __restrict__ k,
                           float* __restrict__ sqk) {
    const int lane = threadIdx.x & 31;
    const int half = lane >> 4, l = lane & 15;
    const int i0 = blockIdx.x * 16, j0 = blockIdx.y * 16;
    const int bh = blockIdx.z, b = bh / NH_, h = bh % NH_;
    const float* Arow = q + (size_t)(b * S_ + i0 + l) * H_ + h * DH_;
    const float* Bcol = k + (size_t)(b * S_ + j0 + l) * H_ + h * DH_; // B[d,j]=k[b,j,h,d]
    v8f acc = {0,0,0,0,0,0,0,0};
    for (int kk = 0; kk < DH_; kk += 4) {
        const int ka = kk + 2 * half;
        acc = wmma4(Arow[ka], Arow[ka + 1], Bcol[ka], Bcol[ka + 1], acc);
    }
#pragma unroll
    for (int r = 0; r < 8; ++r)
        sqk[((size_t)bh * S_ + i0 + half * 8 + r) * S_ + j0 + l] = acc[r];
}

// ------- Kernel 3b: per (b,i): s_edge[12x160] = qe[16x64] x edge[b,i]^T[64x160]
// CDNA5 path: one TDM tensor_load_to_lds stages the whole 40KB edge[b,i] slice
// into LDS (sequential DRAM traffic), with pad_interval=64DW / pad_amount=1DW so
// the LDS row stride is 65 dwords -> lane-strided reads hit 16 distinct banks.
// Block = 320 threads (10 waves); wave w computes j-tile w.
// The TDM intrinsic carries no pointer to smem, so alias analysis cannot see
// the write; an asm escape + memory clobber keeps the ds_loads alive.
__global__ void sedge_kernel(const float* __restrict__ qe, const float* __restrict__ edge,
                             float* __restrict__ sedge) {
    __shared__ float smem[S_ * LDSROW_];          // 160*65*4 = 41600 B
    const int m = blockIdx.x;                     // b*S+i
    const int wave = threadIdx.x >> 5;
    const int lane = threadIdx.x & 31;
    const int half = lane >> 4, l = lane & 15;

    if (wave == 0) {
        // ---- Build Tensor DMA Descriptor (D#) for a 2-D f32 tile 64 x 160 ----
        const unsigned long long ga =
            (unsigned long long)(const void*)(edge + (size_t)m * S_ * DEP_);
        v4u g0;
        g0.x = 1u;                                           // count=1, user mode
        g0.y = 0u;                                           // lds_addr = 0 (smem base)
        g0.z = (unsigned)(ga & 0xFFFFFFFFu);                 // global_addr[31:0]
        g0.w = (unsigned)((ga >> 32) & 0x1FFFFFFu) | (2u << 30); // ga[56:32], type=2
        v8i g1;
        g1[0] = (2 << 16)      // data_size = 4 bytes
              | (1 << 20)      // pad_enable
              | (5 << 22);     // pad_interval: every 64 DWORDs (+ pad_amount=0 -> 1 DWORD)
        g1[1] = (int)(64u << 16);    // tensor_dim0 = 64   (bits 79:48, low16 here)
        g1[2] = (int)(160u << 16);   // tensor_dim1 = 160  (bits 111:80, low16 here)
        g1[3] = (int)(64u << 16);    // tile_dim0 = 64     (bits 127:112)
        g1[4] = 160;                 // tile_dim1 = 160    (bits 143:128)
        g1[5] = 64;                  // tensor_dim0_stride = 64 (bits 207:160, low32)
        g1[6] = (int)((10240u & 0xFFFFu) << 16); // tensor_dim1_stride low16 (bits 223:208)
        g1[7] = 0;                   // tensor_dim1_stride high bits
        v4i gz = {0, 0, 0, 0};
#if defined(__clang_major__) && (__clang_major__ >= 23)
        v8i gz8 = {0, 0, 0, 0, 0, 0, 0, 0};
        __builtin_amdgcn_tensor_load_to_lds(g0, g1, gz, gz, gz8, 0);
#else
        __builtin_amdgcn_tensor_load_to_lds(g0, g1, gz, gz, 0);
#endif
        __builtin_amdgcn_s_wait_tensorcnt(0);
    }
    __syncthreads();
    // Escape smem + memory clobber: the TDM engine wrote LDS behind the
    // compiler's back; without this the smem loads fold to poison.
    {
        const float* esc = &smem[0];
        asm volatile("" :: "v"(esc) : "memory");
    }

    const int j0 = wave * 16;
    const float* Arow = qe + (size_t)m * (16 * DEP_) + l * DEP_;   // A[h=l, dep]
    const int bbase = (j0 + l) * LDSROW_;                          // B[dep, j=j0+l]
    v8f acc = {0,0,0,0,0,0,0,0};
    for (int kk = 0; kk < DEP_; kk += 4) {
        const int ka = kk + 2 * half;
        acc = wmma4(Arow[ka], Arow[ka + 1], smem[bbase + ka], smem[bbase + ka + 1], acc);
    }
#pragma unroll
    for (int r = 0; r < 8; ++r) {
        const int h = half * 8 + r;
        if (h < NH_)
            sedge[((size_t)m * NH_ + h) * S_ + j0 + l] = acc[r];
    }
}

// ---------------- Kernel 4: mask + softmax; probs written over sqk ------------
__global__ void softmax_kernel(const float* __restrict__ sedge,
                               const float* __restrict__ qdb,
                               const int* __restrict__ mask,
                               float* __restrict__ sqk) {
    const int r = blockIdx.x;                 // ((b*NH+h)*S + i)
    const int lane = threadIdx.x;
    const int b = r / (NH_ * S_);
    const int h = (r / S_) % NH_;
    const int i = r % S_;
    float* row = sqk + (size_t)r * S_;
    const float* erow = sedge + ((size_t)(b * S_ + i) * NH_ + h) * S_;
    const int* mrow = mask + (size_t)(b * S_ + i) * S_;
    const float qb = qdb[(size_t)(b * S_ + i) * NH_ + h];
    float vals[5];
    float mx = -3.0e38f;
#pragma unroll
    for (int t = 0; t < 5; ++t) {
        const int j = lane + 32 * t;
        float s = (row[j] + erow[j] + qb) * 0.125f;   // W_EDGE=1, /sqrt(64)
        if (mrow[j] == 0) s = -10000.0f;
        vals[t] = s;
        mx = fmaxf(mx, s);
    }
    for (int off = 16; off; off >>= 1) mx = fmaxf(mx, __shfl_xor(mx, off, 32));
    float sum = 0.0f;
#pragma unroll
    for (int t = 0; t < 5; ++t) { vals[t] = __expf(vals[t] - mx); sum += vals[t]; }
    for (int off = 16; off; off >>= 1) sum += __shfl_xor(sum, off, 32);
    const float inv = 1.0f / sum;
#pragma unroll
    for (int t = 0; t < 5; ++t) row[lane + 32 * t] = vals[t] * inv;
}

// ---------------- Kernel 5a: ctx = probs @ v  (per (b,h): 160x64, K=160) ------
__global__ void ctxv_kernel(const float* __restrict__ probs, const float* __restrict__ v,
                            float* __restrict__ ctx) {
    const int lane = threadIdx.x & 31;
    const int half = lane >> 4, l = lane & 15;
    const int i0 = blockIdx.x * 16, n0 = blockIdx.y * 16;
    const int bh = blockIdx.z, b = bh / NH_, h = bh % NH_;
    const float* Arow = probs + ((size_t)bh * S_ + i0 + l) * S_;
    const float* Bb = v + (size_t)(b * S_) * H_ + h * DH_;      // B[j,d]=v[b,j,h,d]
    v8f acc = {0,0,0,0,0,0,0,0};
    for (int kk = 0; kk < S_; kk += 4) {
        const int ka = kk + 2 * half;
        acc = wmma4(Arow[ka], Arow[ka + 1],
                    Bb[(size_t)ka * H_ + n0 + l], Bb[(size_t)(ka + 1) * H_ + n0 + l],
                    acc);
    }
#pragma unroll
    for (int r = 0; r < 8; ++r)
        ctx[(size_t)(b * S_ + i0 + half * 8 + r) * H_ + h * DH_ + n0 + l] = acc[r];
}

// ------- Kernel 5b: per (b,i): pe[12x64] = probs[12(pad16)x160] x edge[160x64]
// B accesses are already coalesced (lanes walk consecutive dep) -> direct loads.
__global__ void pe_kernel(const float* __restrict__ probs, const float* __restrict__ edge,
                          float* __restrict__ pe) {
    const int lane = threadIdx.x & 31;
    const int half = lane >> 4, l = lane & 15;
    const int m = blockIdx.x;                 // b*S+i
    const int n0 = blockIdx.y * 16;           // dep tile
    const int b = m / S_, i = m % S_;
    const int hh = (l < NH_) ? l : (NH_ - 1); // keep pad rows in-bounds
    const float* Arow = probs + ((size_t)(b * NH_ + hh) * S_ + i) * S_;
    const float* Bb = edge + (size_t)m * S_ * DEP_;   // B[j,dep]=edge[m,j,dep]
    v8f acc = {0,0,0,0,0,0,0,0};
    for (int kk = 0; kk < S_; kk += 4) {
        const int ka = kk + 2 * half;
        acc = wmma4(Arow[ka], Arow[ka + 1],
                    Bb[(size_t)ka * DEP_ + n0 + l], Bb[(size_t)(ka + 1) * DEP_ + n0 + l],
                    acc);
    }
#pragma unroll
    for (int r = 0; r < 8; ++r) {
        const int h = half * 8 + r;
        if (h < NH_)
            pe[(size_t)m * (NH_ * DEP_) + h * DEP_ + n0 + l] = acc[r];
    }
}

// ------- Kernel 5c: ctx += pe_h @ Wev_h + bev_h  (per head: M=1280,N=64,K=64) -
__global__ void ctxe_kernel(const float* __restrict__ pe, const float* __restrict__ Wev,
                            const float* __restrict__ bev, float* __restrict__ ctx) {
    const int lane = threadIdx.x & 31;
    const int half = lane >> 4, l = lane & 15;
    const int m0 = blockIdx.x * 16;
    const int n0 = blockIdx.y * 16;
    const int h  = blockIdx.z;
    const float* Arow = pe + (size_t)(m0 + l) * (NH_ * DEP_) + h * DEP_;
    const float* Bb = Wev + h * DH_;          // B[dep,d] = Wev[dep, h*64+d]
    v8f acc = {0,0,0,0,0,0,0,0};
    for (int kk = 0; kk < DEP_; kk += 4) {
        const int ka = kk + 2 * half;
        acc = wmma4(Arow[ka], Arow[ka + 1],
                    Bb[(size_t)ka * H_ + n0 + l], Bb[(size_t)(ka + 1) * H_ + n0 + l],
                    acc);
    }
    const int col = h * DH_ + n0 + l;
    const float bb = bev[col];
#pragma unroll
    for (int r = 0; r < 8; ++r)
        ctx[(size_t)(m0 + half * 8 + r) * H_ + col] += acc[r] + bb;
}

// ---------------- Kernel 6: out = LayerNorm(token + ctx) ----------------------
__global__ void ln_kernel(const float* __restrict__ X, const float* __restrict__ ctx,
                          const float* __restrict__ g, const float* __restrict__ be,
                          float* __restrict__ out) {
    const int row = blockIdx.x, tid = threadIdx.x;   // 256 threads, 3 elems each
    __shared__ float s1[256], s2[256];
    const size_t base = (size_t)row * H_;
    float x0 = X[base + tid]        + ctx[base + tid];
    float x1 = X[base + tid + 256]  + ctx[base + tid + 256];
    float x2 = X[base + tid + 512]  + ctx[base + tid + 512];
    s1[tid] = x0 + x1 + x2;
    s2[tid] = x0 * x0 + x1 * x1 + x2 * x2;
    __syncthreads();
    for (int off = 128; off; off >>= 1) {
        if (tid < off) { s1[tid] += s1[tid + off]; s2[tid] += s2[tid + off]; }
        __syncthreads();
    }
    const float mean = s1[0] * (1.0f / H_);
    const float var  = s2[0] * (1.0f / H_) - mean * mean;
    const float inv  = rsqrtf(var + 1e-5f);
    out[base + tid]       = (x0 - mean) * inv * g[tid]       + be[tid];
    out[base + tid + 256] = (x1 - mean) * inv * g[tid + 256] + be[tid + 256];
    out[base + tid + 512] = (x2 - mean) * inv * g[tid + 512] + be[tid + 512];
}

extern "C" void kernel_launch(void* const* d_in, const int* in_sizes, int n_in,
                              void* d_out, int out_size, void* d_ws, size_t ws_size,
                              hipStream_t stream) {
    const float* token = (const float*)d_in[0];
    const float* edge  = (const float*)d_in[1];
    const int*   dmask = (const int*)d_in[2];
    const float* Wq = (const float*)d_in[3],  *bq  = (const float*)d_in[4];
    const float* Wk = (const float*)d_in[5],  *bk  = (const float*)d_in[6];
    const float* Wv = (const float*)d_in[7],  *bv  = (const float*)d_in[8];
    const float* Wek = (const float*)d_in[9], *bek = (const float*)d_in[10];
    const float* Wev = (const float*)d_in[11],*bev = (const float*)d_in[12];
    const float* lng = (const float*)d_in[13],*lnb = (const float*)d_in[14];
    float* out = (float*)d_out;

    float* ws = (float*)d_ws;
    size_t off = 0;
    float* q     = ws + off; off += (size_t)M_ * H_;          // 983040
    float* kbuf  = ws + off; off += (size_t)M_ * H_;
    float* vbuf  = ws + off; off += (size_t)M_ * H_;
    float* qe    = ws + off; off += (size_t)M_ * 16 * DEP_;   // head-padded to 16
    float* qdb   = ws + off; off += (size_t)M_ * NH_;
    float* sqk   = ws + off; off += (size_t)B_ * NH_ * S_ * S_;  // reused as probs
    float* sedge = ws + off; off += (size_t)M_ * NH_ * S_;
    float* pe    = ws + off; off += (size_t)M_ * NH_ * DEP_;
    float* ctx   = ws + off; off += (size_t)M_ * H_;

    qkv_kernel   <<<dim3(M_/16, H_/64, 3), 128, 0, stream>>>(token, Wq, bq, Wk, bk, Wv, bv,
                                                             q, kbuf, vbuf);
    qe_kernel    <<<dim3(M_/16, DEP_/16, NH_), 32, 0, stream>>>(q, Wek, qe);
    qbek_kernel  <<<dim3(M_, NH_), 32, 0, stream>>>(q, bek, qdb);
    sqk_kernel   <<<dim3(S_/16, S_/16, B_*NH_), 32, 0, stream>>>(q, kbuf, sqk);
    sedge_kernel <<<dim3(M_), S_/16 * 32, 0, stream>>>(qe, edge, sedge);
    softmax_kernel<<<dim3(B_*NH_*S_), 32, 0, stream>>>(sedge, qdb, dmask, sqk);
    ctxv_kernel  <<<dim3(S_/16, DH_/16, B_*NH_), 32, 0, stream>>>(sqk, vbuf, ctx);
    pe_kernel    <<<dim3(M_, DEP_/16), 32, 0, stream>>>(sqk, edge, pe);
    ctxe_kernel  <<<dim3(M_/16, DH_/16, NH_), 32, 0, stream>>>(pe, Wev, bev, ctx);
    ln_kernel    <<<dim3(M_), 256, 0, stream>>>(token, ctx, lng, lnb, out);
}